// Spherical_sal_loss_71193377898672
// MI455X (gfx1250) — compile-verified
//
#include <hip/hip_runtime.h>
#include <hip/hip_bf16.h>

#define EPS 1e-07f
#define NB 16            // batches
#define NELEM 524288     // 512*1024 elements per batch (C==1)
#define N4 (NELEM / 4)
#define BPB 64           // blocks per batch for both streaming passes
#define IT1 8            // float4 iterations per thread: 131072/(64*256) = 8

// workspace layout in floats
#define P1_OFF 0                        // pass1 block partials: [16*64][8] = 8192
#define M_OFF  8192                     // moments: [16][8] = 128
#define P2_OFF 8320                     // pass2 block partials: [16*64] = 1024
// total = 9344 floats (~37 KB)

typedef __attribute__((ext_vector_type(2))) float v2f;
typedef __attribute__((ext_vector_type(4))) float v4f;
typedef __attribute__((ext_vector_type(8))) float v8f;

#if defined(__has_builtin)
#if __has_builtin(__builtin_amdgcn_wmma_f32_16x16x4_f32)
#define HAVE_WMMA_F32X4 1
#endif
#endif

// Block-wide sum of 8 lane-private accumulators. 256 threads = 8 waves.
// Result valid on wave 0 (use tid==0). Cross-wave stage is one
// V_WMMA_F32_16X16X4_F32 with an all-ones B matrix: D[m][n] = sum_k A[m][k].
//   A[row][k], row = q + 8*wg : wave (k + 4*wg)'s partial for quantity q.
//   D vgpr i = row i (lanes 0-15) and row i+8 (lanes 16-31)
//   => total_q = d[q] + shfl_xor(d[q], 16).
__device__ __forceinline__ void block_reduce8(float p[8], float tot[8]) {
  __shared__ float wsum[8][8];
  const int tid  = threadIdx.x;
  const int lane = tid & 31;
  const int wave = tid >> 5;

#pragma unroll
  for (int q = 0; q < 8; ++q) {
    float v = p[q];
    v += __shfl_xor(v, 1, 32);
    v += __shfl_xor(v, 2, 32);
    v += __shfl_xor(v, 4, 32);
    v += __shfl_xor(v, 8, 32);
    v += __shfl_xor(v, 16, 32);
    p[q] = v;
  }
  if (lane == 0) {
#pragma unroll
    for (int q = 0; q < 8; ++q) wsum[wave][q] = p[q];
  }
  __syncthreads();

  if (wave == 0) {  // uniform branch: EXEC all ones within the wave (WMMA requirement)
#if defined(HAVE_WMMA_F32X4)
    const int row  = lane & 15;
    const int half = lane >> 4;   // 0: K=0,1   1: K=2,3
    const int q    = row & 7;
    const int wg   = row >> 3;    // rows 0-7 -> waves 0-3, rows 8-15 -> waves 4-7
    v2f a;
    a[0] = wsum[2 * half + 0 + 4 * wg][q];   // A[row][2*half]
    a[1] = wsum[2 * half + 1 + 4 * wg][q];   // A[row][2*half+1]
    v2f bones; bones[0] = 1.0f; bones[1] = 1.0f;
    v8f c = {};
    v8f d = __builtin_amdgcn_wmma_f32_16x16x4_f32(
        /*neg_a=*/false, a, /*neg_b=*/false, bones,
        /*c_mod=*/(short)0, c, /*reuse_a=*/false, /*reuse_b=*/false);
#pragma unroll
    for (int i = 0; i < 8; ++i) tot[i] = d[i] + __shfl_xor(d[i], 16, 32);
#else
#pragma unroll
    for (int i = 0; i < 8; ++i) {
      float acc = 0.0f;
#pragma unroll
      for (int w = 0; w < 8; ++w) acc += wsum[w][i];
      tot[i] = acc;
    }
#endif
  }
  __syncthreads();  // allow safe reuse of wsum by subsequent calls
}

// Pass 1: per-batch moments. grid = (64, 16), block = 256.
// Each thread: 8 float4 tiles -> 32 elements. 7 partial sums (+1 pad).
__global__ void pass1_moments(const float* __restrict__ inp,
                              const float* __restrict__ sal,
                              const float* __restrict__ fix,
                              const float* __restrict__ wgt,
                              float* __restrict__ ws) {
  const int blk = blockIdx.x;   // 0..63
  const int b   = blockIdx.y;   // 0..15
  const int tid = threadIdx.x;

  const v4f* x4 = (const v4f*)(inp + (size_t)b * NELEM);
  const v4f* s4 = (const v4f*)(sal + (size_t)b * NELEM);
  const v4f* f4 = (const v4f*)(fix + (size_t)b * NELEM);
  const v4f* w4 = (const v4f*)wgt;

  float p[8];
#pragma unroll
  for (int q = 0; q < 8; ++q) p[q] = 0.0f;

#pragma unroll
  for (int it = 0; it < IT1; ++it) {
    const int i = blk * (256 * IT1) + it * 256 + tid;   // < N4
    const v4f xv = x4[i];
    const v4f sv = s4[i];
    const v4f fv = __builtin_nontemporal_load(&f4[i]);  // fix is read exactly once
    const v4f wv = w4[i];
#pragma unroll
    for (int e = 0; e < 4; ++e) {
      const float x = xv[e] * wv[e];
      const float s = sv[e] * wv[e];
      const float f = fv[e] * wv[e];
      p[0] += x;
      p[1] += x * x;
      p[2] += s;
      p[3] += s * s;
      p[4] += x * s;
      p[5] += f;
      p[6] += x * f;
    }
  }

  float tot[8];
  block_reduce8(p, tot);
  if (tid == 0) {
    float* dst = ws + P1_OFF + ((size_t)(b * BPB + blk)) * 8;
#pragma unroll
    for (int q = 0; q < 8; ++q) dst[q] = tot[q];
  }
}

// Pass 1b: deterministically fold 64 block partials per batch. grid = 16.
__global__ void pass1_reduce(float* __restrict__ ws) {
  const int b   = blockIdx.x;
  const int tid = threadIdx.x;
  float p[8];
#pragma unroll
  for (int q = 0; q < 8; ++q) p[q] = 0.0f;
  if (tid < BPB) {
    const float* src = ws + P1_OFF + ((size_t)(b * BPB + tid)) * 8;
#pragma unroll
    for (int q = 0; q < 8; ++q) p[q] = src[q];
  }
  float tot[8];
  block_reduce8(p, tot);
  if (tid == 0) {
#pragma unroll
    for (int q = 0; q < 8; ++q) ws[M_OFF + b * 8 + q] = tot[q];
  }
}

// Pass 2: KLD elementwise pass. grid = (64, 16), block = 256.
// Each thread: 8 float4 tiles -> 32 elements. Re-reads x,s (expected L2-resident).
__global__ void pass2_kld(const float* __restrict__ inp,
                          const float* __restrict__ sal,
                          const float* __restrict__ wgt,
                          float* __restrict__ ws) {
  const int blk = blockIdx.x;   // 0..63
  const int b   = blockIdx.y;
  const int tid = threadIdx.x;

  const float Sx = ws[M_OFF + b * 8 + 0];
  const float Ss = ws[M_OFF + b * 8 + 2];

  const v4f* x4 = (const v4f*)(inp + (size_t)b * NELEM);
  const v4f* s4 = (const v4f*)(sal + (size_t)b * NELEM);
  const v4f* w4 = (const v4f*)wgt;

  float acc = 0.0f;
#pragma unroll
  for (int it = 0; it < IT1; ++it) {
    const int i = blk * (256 * IT1) + it * 256 + tid;  // < N4
    const v4f xv = x4[i];
    const v4f sv = s4[i];
    const v4f wv = w4[i];
#pragma unroll
    for (int e = 0; e < 4; ++e) {
      const float xn = (xv[e] * wv[e]) / Sx;
      const float sn = (sv[e] * wv[e]) / Ss;
      acc += sn * logf(sn / (xn + EPS) + EPS);
    }
  }

  float p[8];
  p[0] = acc;
#pragma unroll
  for (int q = 1; q < 8; ++q) p[q] = 0.0f;
  float tot[8];
  block_reduce8(p, tot);
  if (tid == 0) ws[P2_OFF + b * BPB + blk] = tot[0];
}

// Finalize: fold KLD partials per batch, evaluate NSS/CC from moments, emit scalar.
__global__ void finalize(const float* __restrict__ wsc, float* __restrict__ out) {
  const int tid = threadIdx.x;
  const float n = (float)NELEM;

  float nss_sum = 0.0f, kld_sum = 0.0f, cc_sum = 0.0f;
  for (int b = 0; b < NB; ++b) {
    float p[8];
#pragma unroll
    for (int q = 0; q < 8; ++q) p[q] = 0.0f;
    if (tid < BPB) p[0] = wsc[P2_OFF + b * BPB + tid];
    float tot[8];
    block_reduce8(p, tot);
    if (tid == 0) {
      const float kld_b = tot[0];
      const float* m = wsc + M_OFF + b * 8;
      const float Sx = m[0], Sx2 = m[1], Ss = m[2], Ss2 = m[3];
      const float Sxs = m[4], Sf = m[5], Sxf = m[6];
      // NSS: ((Sxf - mu*Sf)/(sd+eps)) / (Sf+eps), sd with ddof=1
      const float mu  = Sx / n;
      const float var = (Sx2 - Sx * Sx / n) / (n - 1.0f);
      const float sd  = sqrtf(var);
      const float nss_b = (Sxf - mu * Sf) / ((sd + EPS) * (Sf + EPS));
      // CC on normalized x,s: sums become 1, second moments scale by 1/S^2
      const float num = Sxs / (Sx * Ss) - 1.0f / n;
      const float den = sqrtf((Sx2 / (Sx * Sx) - 1.0f / n) *
                              (Ss2 / (Ss * Ss) - 1.0f / n));
      const float cc_b = num / (den + EPS);
      nss_sum += nss_b;
      kld_sum += kld_b;
      cc_sum  += cc_b;
    }
  }
  if (tid == 0) {
    const float invB = 1.0f / (float)NB;
    out[0] = -0.1f * (nss_sum * invB) + (kld_sum * invB) - 0.1f * (cc_sum * invB);
  }
}

extern "C" void kernel_launch(void* const* d_in, const int* in_sizes, int n_in,
                              void* d_out, int out_size, void* d_ws, size_t ws_size,
                              hipStream_t stream) {
  const float* inp = (const float*)d_in[0];
  const float* sal = (const float*)d_in[1];
  const float* fix = (const float*)d_in[2];
  const float* wgt = (const float*)d_in[3];
  float* ws  = (float*)d_ws;
  float* out = (float*)d_out;

  // Every workspace slot is written before it is read each call: no init needed,
  // no float atomics anywhere -> bitwise-deterministic across graph replays.
  pass1_moments<<<dim3(BPB, NB), 256, 0, stream>>>(inp, sal, fix, wgt, ws);
  pass1_reduce<<<NB, 256, 0, stream>>>(ws);
  pass2_kld<<<dim3(BPB, NB), 256, 0, stream>>>(inp, sal, wgt, ws);
  finalize<<<1, 256, 0, stream>>>(ws, out);
}